// MultiGraphConvolution_Layer_87771951661826
// MI455X (gfx1250) — compile-verified
//
#include <hip/hip_runtime.h>
#include <math.h>

#define N_NODES 50000
#define N_EDGES 800000
#define D 128
#define TOT_E (N_EDGES + N_NODES)
#define NEG_SLOPE 0.2f

typedef __attribute__((ext_vector_type(2))) float v2f;
typedef __attribute__((ext_vector_type(8))) float v8f;

// ---- monotone float<->u32 order map (for atomic max on float via u32) ----
__device__ __forceinline__ unsigned ford(float f) {
  unsigned u = __float_as_uint(f);
  return (u & 0x80000000u) ? ~u : (u | 0x80000000u);
}
__device__ __forceinline__ float funord(unsigned k) {
  unsigned u = (k & 0x80000000u) ? (k ^ 0x80000000u) : ~k;
  return __uint_as_float(u);
}

__device__ __forceinline__ void atomAddF(float* p, float v) {
  __hip_atomic_fetch_add(p, v, __ATOMIC_RELAXED, __HIP_MEMORY_SCOPE_AGENT);
}
__device__ __forceinline__ void atomMaxU(unsigned* p, unsigned v) {
  __hip_atomic_fetch_max(p, v, __ATOMIC_RELAXED, __HIP_MEMORY_SCOPE_AGENT);
}

// ================= H = X @ W  via V_WMMA_F32_16X16X4_F32 ==================
// X: M x 128 row-major, W: 128 x 128 row-major, H: M x 128.
// Block = 256 threads = 8 waves; blockIdx.x = M-tile (M multiple of 16);
// wave w computes columns [16w, 16w+16). Per ISA A-layout (16x4 f32):
// lanes 0-15 hold K=0,1 ; lanes 16-31 hold K=2,3 (M = lane%16). B mirrors
// with N = lane%16. C/D: VGPR v -> row v + 8*(lane>=16), col lane%16.
__global__ __launch_bounds__(256) void gat_gemm_wmma(const float* __restrict__ X,
                                                     const float* __restrict__ W,
                                                     float* __restrict__ H) {
  const int wave = threadIdx.x >> 5;
  const int lane = threadIdx.x & 31;
  const int m0 = blockIdx.x << 4;
  const int n0 = wave << 4;
  const int lh = lane >> 4;     // which K-half this lane holds
  const int l16 = lane & 15;
  v8f acc = {0.f, 0.f, 0.f, 0.f, 0.f, 0.f, 0.f, 0.f};
  const float* xr = X + (size_t)(m0 + l16) * D;
#pragma unroll 4
  for (int k = 0; k < D; k += 4) {
    const int ka = k + (lh << 1);
    v2f a, b;
    a.x = xr[ka];
    a.y = xr[ka + 1];
    b.x = W[(size_t)ka * D + n0 + l16];
    b.y = W[(size_t)(ka + 1) * D + n0 + l16];
    acc = __builtin_amdgcn_wmma_f32_16x16x4_f32(false, a, false, b,
                                                (short)0, acc, false, false);
  }
  float* hp = H + (size_t)(m0 + (lh << 3)) * D + n0 + l16;
#pragma unroll
  for (int v = 0; v < 8; ++v) hp[(size_t)v * D] = acc[v];
}

// ============== per-node alpha_src/alpha_dst = H[n] . a_vec ==============
__global__ __launch_bounds__(256) void gat_alpha(const float* __restrict__ H,
                                                 const float* __restrict__ a_src,
                                                 const float* __restrict__ a_dst,
                                                 float* __restrict__ as,
                                                 float* __restrict__ ad) {
  const int node = blockIdx.x * 8 + (threadIdx.x >> 5);
  const int lane = threadIdx.x & 31;
  if (node >= N_NODES) return;
  const float* h = H + (size_t)node * D;
  float s0 = 0.f, s1 = 0.f;
#pragma unroll
  for (int j = 0; j < D; j += 32) {
    float hv = h[j + lane];
    s0 += hv * a_src[j + lane];
    s1 += hv * a_dst[j + lane];
  }
#pragma unroll
  for (int off = 16; off; off >>= 1) {
    s0 += __shfl_xor(s0, off, 32);
    s1 += __shfl_xor(s1, off, 32);
  }
  if (lane == 0) { as[node] = s0; ad[node] = s1; }
}

// ============== init: m=ordered(-inf)=0, s=0, accumulator=0 ==============
__global__ void gat_init(unsigned* __restrict__ mU, float* __restrict__ s,
                         float* __restrict__ acc) {
  const int t = blockIdx.x * blockDim.x + threadIdx.x;
  if (t < N_NODES) { mU[t] = 0u; s[t] = 0.f; }
  if (t < N_NODES * D) acc[t] = 0.f;
}

// ====== pass 1: e = leaky_relu(as[src]+ad[dst]); segment max into mU ======
__global__ void gat_edge_max(const int* __restrict__ ei,
                             const float* __restrict__ as,
                             const float* __restrict__ ad,
                             float* __restrict__ ebuf,
                             unsigned* __restrict__ mU) {
  const int t = blockIdx.x * blockDim.x + threadIdx.x;
  if (t >= TOT_E) return;
  const int src = (t < N_EDGES) ? ei[t] : (t - N_EDGES);
  const int dst = (t < N_EDGES) ? ei[N_EDGES + t] : (t - N_EDGES);
  float e = as[src] + ad[dst];
  e = (e > 0.f) ? e : NEG_SLOPE * e;
  ebuf[t] = e;
  atomMaxU(&mU[dst], ford(e));
}

// ====== pass 2: ex = exp(e - m[dst]) (in place); s[dst] += ex ======
__global__ void gat_edge_exp(const int* __restrict__ ei,
                             float* __restrict__ ebuf,
                             const unsigned* __restrict__ mU,
                             float* __restrict__ s) {
  const int t = blockIdx.x * blockDim.x + threadIdx.x;
  if (t >= TOT_E) return;
  const int dst = (t < N_EDGES) ? ei[N_EDGES + t] : (t - N_EDGES);
  float m = funord(mU[dst]);
  if (!isfinite(m)) m = 0.f;
  const float ex = expf(ebuf[t] - m);
  ebuf[t] = ex;
  atomAddF(&s[dst], ex);
}

// ====== pass 3 (heavy): acc[dst] += ex * H[src]; one wave per edge ======
__global__ __launch_bounds__(256) void gat_edge_agg(const int* __restrict__ ei,
                                                    const float* __restrict__ H,
                                                    const float* __restrict__ ex,
                                                    float* __restrict__ acc) {
  const int e = blockIdx.x * 8 + (threadIdx.x >> 5);
  const int lane = threadIdx.x & 31;
  if (e >= TOT_E) return;
  const int src = (e < N_EDGES) ? ei[e] : (e - N_EDGES);
  const int dst = (e < N_EDGES) ? ei[N_EDGES + e] : (e - N_EDGES);
  const float w = ex[e];
  const float4 hv = ((const float4*)(H + (size_t)src * D))[lane];
  float* op = acc + (size_t)dst * D + 4 * lane;
  atomAddF(op + 0, w * hv.x);
  atomAddF(op + 1, w * hv.y);
  atomAddF(op + 2, w * hv.z);
  atomAddF(op + 3, w * hv.w);
}

// ====== finalize: out = relu(acc / (s + 1e-16) + b), in place ======
__global__ void gat_finalize(float* __restrict__ acc,
                             const float* __restrict__ s,
                             const float* __restrict__ b) {
  const int t = blockIdx.x * blockDim.x + threadIdx.x;
  if (t >= N_NODES * D) return;
  const int node = t >> 7;
  const int f = t & (D - 1);
  const float v = acc[t] / (s[node] + 1e-16f) + b[f];
  acc[t] = (v > 0.f) ? v : 0.f;
}

// ============================ host side ============================
static void run_layer(const float* Xin, const int* ei, const float* W,
                      const float* av_s, const float* av_d, const float* bv,
                      float* H, float* ebuf, float* as, float* ad,
                      unsigned* mU, float* ssum, float* out,
                      hipStream_t stream) {
  gat_gemm_wmma<<<N_NODES / 16, 256, 0, stream>>>(Xin, W, H);
  gat_alpha<<<(N_NODES + 7) / 8, 256, 0, stream>>>(H, av_s, av_d, as, ad);
  gat_init<<<(N_NODES * D + 255) / 256, 256, 0, stream>>>(mU, ssum, out);
  gat_edge_max<<<(TOT_E + 255) / 256, 256, 0, stream>>>(ei, as, ad, ebuf, mU);
  gat_edge_exp<<<(TOT_E + 255) / 256, 256, 0, stream>>>(ei, ebuf, mU, ssum);
  gat_edge_agg<<<(TOT_E + 7) / 8, 256, 0, stream>>>(ei, H, ebuf, out);
  gat_finalize<<<(N_NODES * D + 255) / 256, 256, 0, stream>>>(out, ssum, bv);
}

extern "C" void kernel_launch(void* const* d_in, const int* in_sizes, int n_in,
                              void* d_out, int out_size, void* d_ws, size_t ws_size,
                              hipStream_t stream) {
  const float* x      = (const float*)d_in[0];
  const int*   ei     = (const int*)d_in[1];   // edge_index [2, E] flat
  const float* W1     = (const float*)d_in[2];
  const float* a_src1 = (const float*)d_in[3];
  const float* a_dst1 = (const float*)d_in[4];
  const float* b1     = (const float*)d_in[5];
  const float* W2     = (const float*)d_in[6];
  const float* a_src2 = (const float*)d_in[7];
  const float* a_dst2 = (const float*)d_in[8];
  const float* b2     = (const float*)d_in[9];
  float* out = (float*)d_out;

  // workspace layout (floats)
  float* ws = (float*)d_ws;
  float*    H    = ws;                          // N*D
  float*    H1   = H + (size_t)N_NODES * D;     // N*D   (layer-1 output)
  float*    ebuf = H1 + (size_t)N_NODES * D;    // TOT_E (e, then ex, in place)
  float*    as   = ebuf + TOT_E;                // N
  float*    ad   = as + N_NODES;                // N
  float*    ssum = ad + N_NODES;                // N
  unsigned* mU   = (unsigned*)(ssum + N_NODES); // N

  // layer 1: x -> H1
  run_layer(x, ei, W1, a_src1, a_dst1, b1, H, ebuf, as, ad, mU, ssum, H1, stream);
  // layer 2: H1 -> d_out
  run_layer(H1, ei, W2, a_src2, a_dst2, b2, H, ebuf, as, ad, mU, ssum, out, stream);
}